// Warp_90331752170249
// MI455X (gfx1250) — compile-verified
//
#include <hip/hip_runtime.h>
#include <stdint.h>

// ---------------------------------------------------------------------------
// SoftSplat pyramid forward-warp for MI455X (gfx1250, wave32).
//
// - All dims compile-time (templates) -> no runtime int-div sequences.
// - Scatter uses no-return global_atomic_add_f32 (relaxed, agent scope).
// - Splat channel walk uses CDNA5 async global->LDS copies (ASYNCcnt,
//   double-buffered, s_wait_asynccnt 1) when the builtin exists; otherwise
//   direct coalesced loads + global_prefetch_b8.
// ---------------------------------------------------------------------------

#define EPSF 1e-7f
#define BN   2

typedef __attribute__((address_space(1))) int* as1_ip;   // "__device__ int *"
typedef __attribute__((address_space(3))) int* as3_ip;   // "__shared__ int *"

#if __has_builtin(__builtin_amdgcn_global_load_async_to_lds_b32)
#define HAVE_ASYNC_LDS 1
__device__ __forceinline__ void asyncLoadF32(float* lds, const float* g) {
    // global VA == flat VA, so integer round-trip is exact for AS1;
    // LDS needs a real addrspacecast (flat aperture addr != LDS offset).
    __builtin_amdgcn_global_load_async_to_lds_b32(
        (as1_ip)(uintptr_t)g, (as3_ip)(void*)lds, 0, 0);
}
#else
#define HAVE_ASYNC_LDS 0
#endif

#if __has_builtin(__builtin_amdgcn_s_wait_asynccnt)
#define WAIT_ASYNC(n) __builtin_amdgcn_s_wait_asynccnt(n)
#else
#define WAIT_ASYNC(n) asm volatile("s_wait_asynccnt %0" :: "n"(n) : "memory")
#endif

__device__ __forceinline__ void atomAddF(float* p, float v) {
    // no-return f32 atomic add -> global_atomic_add_f32 (STOREcnt, no wait)
    __hip_atomic_fetch_add(p, v, __ATOMIC_RELAXED, __HIP_MEMORY_SCOPE_AGENT);
}

// PyTorch bilinear source index, align_corners=False (matches _src_idx).
template <int OUT, int IN>
__device__ __forceinline__ void srcIdx(int i, int& i0, int& i1, float& lam) {
    constexpr float scale = (float)IN / (float)OUT;
    float s = ((float)i + 0.5f) * scale - 0.5f;
    s = fmaxf(s, 0.0f);
    float f = floorf(s);
    i0 = (int)f;
    i1 = min(i0 + 1, IN - 1);
    lam = s - f;
}

// (BC,h,w) -> (BC,oh,ow) bilinear gather with uniform scale (flow).
template <int h, int w, int oh, int ow>
__global__ void resize_bilinear_k(const float* __restrict__ src, float* __restrict__ dst,
                                  int BC, float mul) {
    int idx = blockIdx.x * blockDim.x + threadIdx.x;
    if (idx >= BC * oh * ow) return;
    int x  = idx % ow;
    int t  = idx / ow;
    int y  = t % oh;
    int bc = t / oh;
    int y0, y1, x0, x1; float ly, lx;
    srcIdx<oh, h>(y, y0, y1, ly);
    srcIdx<ow, w>(x, x0, x1, lx);
    const float* s = src + (size_t)bc * h * w;
    float r0 = s[y0 * w + x0] * (1.0f - ly) + s[y1 * w + x0] * ly;
    float r1 = s[y0 * w + x1] * (1.0f - ly) + s[y1 * w + x1] * ly;
    dst[idx] = (r0 * (1.0f - lx) + r1 * lx) * mul;
}

__device__ __forceinline__ void scatter4(float* o, long long k, int valid, int Wd,
                                         float val, float w00, float w10,
                                         float w01, float w11) {
    if (valid & 1) atomAddF(o + k,           val * w00);
    if (valid & 2) atomAddF(o + k + 1,       val * w10);
    if (valid & 4) atomAddF(o + k + Wd,      val * w01);
    if (valid & 8) atomAddF(o + k + Wd + 1,  val * w11);
}

// Soft-splat of concat([enc, m]) with metric clip(-m,-20,20); hole splat fused.
// Block = (32 pixels, 8 channel-slices); each wave is one channel-slice.
template <int C, int H, int W>
__global__ __launch_bounds__(256)
void splat_soft_k(const float* __restrict__ enc, const float* __restrict__ m,
                  const float* __restrict__ flow, float* __restrict__ outv,
                  float* __restrict__ wacc, float* __restrict__ hacc) {
    constexpr int HW    = H * W;
    constexpr int total = BN * HW;
    const int tx = threadIdx.x, ty = threadIdx.y;
    const int pix = blockIdx.x * 32 + tx;

    __shared__ float sw00[32], sw10[32], sw01[32], sw11[32];
    __shared__ float se[32], smv[32];
    __shared__ int   sbase[32], svalid[32];

    if (ty == 0) {
        float w00 = 0, w10 = 0, w01 = 0, w11 = 0, e = 0, mv = 0;
        int base = 0, valid = 0;
        if (pix < total) {
            int b = pix / HW, p = pix - b * HW;
            int y = p / W,    x = p - y * W;
            float fx = (float)x + flow[((size_t)b * 2 + 0) * HW + p];
            float fy = (float)y + flow[((size_t)b * 2 + 1) * HW + p];
            float x0f = floorf(fx), y0f = floorf(fy);
            int x0 = (int)x0f, y0 = (int)y0f;
            float wx = fx - x0f, wy = fy - y0f;
            w00 = (1.0f - wx) * (1.0f - wy);
            w10 = wx * (1.0f - wy);
            w01 = (1.0f - wx) * wy;
            w11 = wx * wy;
            mv = m[(size_t)b * HW + p];
            e  = expf(fminf(fmaxf(-mv, -20.0f), 20.0f));
            base = y0 * W + x0;
            bool X0 = (x0 >= 0) && (x0 < W);
            bool X1 = (x0 + 1 >= 0) && (x0 + 1 < W);
            bool Y0 = (y0 >= 0) && (y0 < H);
            bool Y1 = (y0 + 1 >= 0) && (y0 + 1 < H);
            valid = (X0 && Y0 ? 1 : 0) | (X1 && Y0 ? 2 : 0) |
                    (X0 && Y1 ? 4 : 0) | (X1 && Y1 ? 8 : 0);
        }
        sw00[tx] = w00; sw10[tx] = w10; sw01[tx] = w01; sw11[tx] = w11;
        se[tx] = e; smv[tx] = mv; sbase[tx] = base; svalid[tx] = valid;
    }
#if HAVE_ASYNC_LDS
    __shared__ float sbuf[2][8][32];   // double buffer per channel-slice wave
#endif
    __syncthreads();
    if (pix >= total) return;
    const int valid = svalid[tx];
    if (valid == 0) return;          // all corners OOB: reference adds exact 0s
    const int b = pix / HW, p = pix - b * HW;
    const float e = se[tx];
    const float mval = smv[tx];
    const float w00 = sw00[tx], w10 = sw10[tx], w01 = sw01[tx], w11 = sw11[tx];
    const long long k = sbase[tx];   // may be negative; only valid corners deref
    const float* encp = enc + (size_t)b * C * HW + p;

    // channels 0..C-1 = enc*e ; channel C = m*e (weight channel lives in wacc)
#if HAVE_ASYNC_LDS
    {
        int c = ty;
        if (c < C) asyncLoadF32(&sbuf[0][ty][tx], encp + (size_t)c * HW);
        int bufi = 0;
        for (; c <= C; c += 8, bufi ^= 1) {
            const int  cn  = c + 8;
            const bool nxt = (cn < C);          // uniform across the wave
            if (nxt) asyncLoadF32(&sbuf[bufi ^ 1][ty][tx], encp + (size_t)cn * HW);
            if (nxt) { WAIT_ASYNC(1); } else { WAIT_ASYNC(0); }
            float val = ((c < C) ? sbuf[bufi][ty][tx] : mval) * e;
            float* o = outv + ((size_t)b * (C + 1) + c) * HW;
            scatter4(o, k, valid, W, val, w00, w10, w01, w11);
        }
    }
#else
    for (int c = ty; c <= C; c += 8) {
        if (c + 8 < C) __builtin_prefetch(encp + (size_t)(c + 8) * HW, 0, 3);
        float val = ((c < C) ? encp[(size_t)c * HW] : mval) * e;
        float* o = outv + ((size_t)b * (C + 1) + c) * HW;
        scatter4(o, k, valid, W, val, w00, w10, w01, w11);
    }
#endif
    if (ty == 0) {
        scatter4(wacc + (size_t)b * HW, k, valid, W, e,    w00, w10, w01, w11);
        scatter4(hacc + (size_t)b * HW, k, valid, W, 1.0f, w00, w10, w01, w11);
    }
}

// Hole-only splat of ones (level-3 masks).
template <int H, int W>
__global__ void splat_hole_k(const float* __restrict__ flow, float* __restrict__ hacc) {
    constexpr int HW = H * W;
    int idx = blockIdx.x * blockDim.x + threadIdx.x;
    if (idx >= BN * HW) return;
    int b = idx / HW, p = idx - b * HW;
    int y = p / W,    x = p - y * W;
    float fx = (float)x + flow[((size_t)b * 2 + 0) * HW + p];
    float fy = (float)y + flow[((size_t)b * 2 + 1) * HW + p];
    float x0f = floorf(fx), y0f = floorf(fy);
    int x0 = (int)x0f, y0 = (int)y0f;
    float wx = fx - x0f, wy = fy - y0f;
    long long k = (long long)y0 * W + x0;
    int valid = ((x0 >= 0 && x0 < W && y0 >= 0 && y0 < H) ? 1 : 0) |
                ((x0 + 1 >= 0 && x0 + 1 < W && y0 >= 0 && y0 < H) ? 2 : 0) |
                ((x0 >= 0 && x0 < W && y0 + 1 >= 0 && y0 + 1 < H) ? 4 : 0) |
                ((x0 + 1 >= 0 && x0 + 1 < W && y0 + 1 >= 0 && y0 + 1 < H) ? 8 : 0);
    scatter4(hacc + (size_t)b * HW, k, valid, W, 1.0f,
             (1.0f - wx) * (1.0f - wy), wx * (1.0f - wy),
             (1.0f - wx) * wy,          wx * wy);
}

// out[:, :C+1] /= (wsum + EPS), in place over d_out.
template <int C1, int HW>
__global__ void finalize_k(float* __restrict__ outv, const float* __restrict__ wacc) {
    constexpr int total = BN * C1 * HW;
    int idx = blockIdx.x * blockDim.x + threadIdx.x;
    if (idx >= total) return;
    int p  = idx % HW;
    int bc = idx / HW;
    int b  = bc / C1;
    outv[idx] = outv[idx] / (wacc[(size_t)b * HW + p] + EPSF);
}

// hole = ( s/(s+EPS) <= 0.5 ) ? 1 : 0
__global__ void hole_k(const float* __restrict__ hacc, float* __restrict__ outm, int n) {
    int idx = blockIdx.x * blockDim.x + threadIdx.x;
    if (idx >= n) return;
    float s = hacc[idx];
    float v = s / (s + EPSF);
    outm[idx] = (v <= 0.5f) ? 1.0f : 0.0f;
}

extern "C" void kernel_launch(void* const* d_in, const int* in_sizes, int n_in,
                              void* d_out, int out_size, void* d_ws, size_t ws_size,
                              hipStream_t stream) {
    // setup_inputs() dict insertion order: enc1_0, enc2_0, enc1_1, enc2_1,
    // enc1_2, enc2_2, metric1, metric2, flow_fwd, flow_bwd
    const float* enc1_0  = (const float*)d_in[0];
    const float* enc2_0  = (const float*)d_in[1];
    const float* enc1_1  = (const float*)d_in[2];
    const float* enc2_1  = (const float*)d_in[3];
    const float* enc1_2  = (const float*)d_in[4];
    const float* enc2_2  = (const float*)d_in[5];
    const float* metric1 = (const float*)d_in[6];
    const float* metric2 = (const float*)d_in[7];
    const float* flow_f  = (const float*)d_in[8];
    const float* flow_b  = (const float*)d_in[9];
    float* out = (float*)d_out;

    const int Hs[4] = {384, 192, 96, 48};
    const int Ws[4] = {640, 320, 160, 80};
    const int Cs[3] = {48, 96, 192};

    // ---- carve d_out: outs (pairs per level), then masks (pairs, 4 levels) ----
    float* outT[6];
    float* maskT[8];
    {
        size_t off = 0;
        for (int l = 0; l < 3; l++) {
            size_t s = (size_t)BN * (Cs[l] + 1) * Hs[l] * Ws[l];
            outT[2 * l] = out + off;     off += s;
            outT[2 * l + 1] = out + off; off += s;
        }
        for (int l = 0; l < 4; l++) {
            size_t s = (size_t)BN * Hs[l] * Ws[l];
            maskT[2 * l] = out + off;     off += s;
            maskT[2 * l + 1] = out + off; off += s;
        }
    }

    // ---- carve workspace: [wacc | hacc] (zeroed), then resize buffers ----
    float* ws = (float*)d_ws;
    size_t wo = 0;
    float* wacc[6];
    float* hacc[8];
    for (int l = 0; l < 3; l++) {
        size_t s = (size_t)BN * Hs[l] * Ws[l];
        wacc[2 * l] = ws + wo; wo += s;
        wacc[2 * l + 1] = ws + wo; wo += s;
    }
    for (int l = 0; l < 4; l++) {
        size_t s = (size_t)BN * Hs[l] * Ws[l];
        hacc[2 * l] = ws + wo; wo += s;
        hacc[2 * l + 1] = ws + wo; wo += s;
    }
    const size_t zeroF = wo;
    float *m1r[2], *m2r[2];   // resized metrics at levels 1,2
    for (int i = 0; i < 2; i++) {
        int l = i + 1;
        size_t s = (size_t)BN * Hs[l] * Ws[l];
        m1r[i] = ws + wo; wo += s;
        m2r[i] = ws + wo; wo += s;
    }
    float *ffr[3], *fbr[3];   // resized flows at levels 1,2,3
    for (int i = 0; i < 3; i++) {
        int l = i + 1;
        size_t s = (size_t)BN * 2 * Hs[l] * Ws[l];
        ffr[i] = ws + wo; wo += s;
        fbr[i] = ws + wo; wo += s;
    }
    (void)ws_size; (void)in_sizes; (void)n_in;

    (void)hipMemsetAsync(d_out, 0, (size_t)out_size * sizeof(float), stream);
    (void)hipMemsetAsync(d_ws, 0, zeroF * sizeof(float), stream);

    const int TPB = 256;
    auto g1 = [](size_t n) { return dim3((unsigned)((n + 255) / 256)); };
    auto gs = [](size_t n) { return dim3((unsigned)((n + 31) / 32)); };
    dim3 sblk(32, 8);

    // ---- progressive resizes (faithful chain: level k built from level k-1) ----
    // level 1: from originals; flow scale = 192/384 = 0.5
    resize_bilinear_k<384, 640, 192, 320><<<g1((size_t)BN * 1 * 192 * 320), TPB, 0, stream>>>(
        metric1, m1r[0], BN * 1, 1.0f);
    resize_bilinear_k<384, 640, 192, 320><<<g1((size_t)BN * 1 * 192 * 320), TPB, 0, stream>>>(
        metric2, m2r[0], BN * 1, 1.0f);
    resize_bilinear_k<384, 640, 192, 320><<<g1((size_t)BN * 2 * 192 * 320), TPB, 0, stream>>>(
        flow_f, ffr[0], BN * 2, 0.5f);
    resize_bilinear_k<384, 640, 192, 320><<<g1((size_t)BN * 2 * 192 * 320), TPB, 0, stream>>>(
        flow_b, fbr[0], BN * 2, 0.5f);
    // level 2: from level 1; flow scale = 96/192 = 0.5
    resize_bilinear_k<192, 320, 96, 160><<<g1((size_t)BN * 1 * 96 * 160), TPB, 0, stream>>>(
        m1r[0], m1r[1], BN * 1, 1.0f);
    resize_bilinear_k<192, 320, 96, 160><<<g1((size_t)BN * 1 * 96 * 160), TPB, 0, stream>>>(
        m2r[0], m2r[1], BN * 1, 1.0f);
    resize_bilinear_k<192, 320, 96, 160><<<g1((size_t)BN * 2 * 96 * 160), TPB, 0, stream>>>(
        ffr[0], ffr[1], BN * 2, 0.5f);
    resize_bilinear_k<192, 320, 96, 160><<<g1((size_t)BN * 2 * 96 * 160), TPB, 0, stream>>>(
        fbr[0], fbr[1], BN * 2, 0.5f);
    // level 3 (masks only): from level 2; flow scale = (96//2)/192 = 0.25 (faithful)
    resize_bilinear_k<96, 160, 48, 80><<<g1((size_t)BN * 2 * 48 * 80), TPB, 0, stream>>>(
        ffr[1], ffr[2], BN * 2, 0.25f);
    resize_bilinear_k<96, 160, 48, 80><<<g1((size_t)BN * 2 * 48 * 80), TPB, 0, stream>>>(
        fbr[1], fbr[2], BN * 2, 0.25f);

    // ---- soft splats (hole splat fused), per level, both directions ----
    splat_soft_k<48, 384, 640><<<gs((size_t)BN * 384 * 640), sblk, 0, stream>>>(
        enc1_0, metric1, flow_f, outT[0], wacc[0], hacc[0]);
    splat_soft_k<48, 384, 640><<<gs((size_t)BN * 384 * 640), sblk, 0, stream>>>(
        enc2_0, metric2, flow_b, outT[1], wacc[1], hacc[1]);
    splat_soft_k<96, 192, 320><<<gs((size_t)BN * 192 * 320), sblk, 0, stream>>>(
        enc1_1, m1r[0], ffr[0], outT[2], wacc[2], hacc[2]);
    splat_soft_k<96, 192, 320><<<gs((size_t)BN * 192 * 320), sblk, 0, stream>>>(
        enc2_1, m2r[0], fbr[0], outT[3], wacc[3], hacc[3]);
    splat_soft_k<192, 96, 160><<<gs((size_t)BN * 96 * 160), sblk, 0, stream>>>(
        enc1_2, m1r[1], ffr[1], outT[4], wacc[4], hacc[4]);
    splat_soft_k<192, 96, 160><<<gs((size_t)BN * 96 * 160), sblk, 0, stream>>>(
        enc2_2, m2r[1], fbr[1], outT[5], wacc[5], hacc[5]);
    // level-3 hole-only splats
    splat_hole_k<48, 80><<<g1((size_t)BN * 48 * 80), TPB, 0, stream>>>(ffr[2], hacc[6]);
    splat_hole_k<48, 80><<<g1((size_t)BN * 48 * 80), TPB, 0, stream>>>(fbr[2], hacc[7]);

    // ---- normalize outs in place; binarize hole masks ----
    finalize_k<49, 384 * 640><<<g1((size_t)BN * 49 * 384 * 640), TPB, 0, stream>>>(outT[0], wacc[0]);
    finalize_k<49, 384 * 640><<<g1((size_t)BN * 49 * 384 * 640), TPB, 0, stream>>>(outT[1], wacc[1]);
    finalize_k<97, 192 * 320><<<g1((size_t)BN * 97 * 192 * 320), TPB, 0, stream>>>(outT[2], wacc[2]);
    finalize_k<97, 192 * 320><<<g1((size_t)BN * 97 * 192 * 320), TPB, 0, stream>>>(outT[3], wacc[3]);
    finalize_k<193, 96 * 160><<<g1((size_t)BN * 193 * 96 * 160), TPB, 0, stream>>>(outT[4], wacc[4]);
    finalize_k<193, 96 * 160><<<g1((size_t)BN * 193 * 96 * 160), TPB, 0, stream>>>(outT[5], wacc[5]);
    for (int i = 0; i < 8; i++) {
        int l = i / 2;
        int n = BN * Hs[l] * Ws[l];
        hole_k<<<g1((size_t)n), TPB, 0, stream>>>(hacc[i], maskT[i], n);
    }
}